// DisplacementLoss_21483426415144
// MI455X (gfx1250) — compile-verified
//
#include <hip/hip_runtime.h>
#include <math.h>

typedef __attribute__((ext_vector_type(2))) float v2f;
typedef __attribute__((ext_vector_type(8))) float v8f;

#define TILE 16

// Zero the scalar output accumulator.
__global__ void chamfer_zero_kernel(float* out, int n) {
    int i = blockIdx.x * blockDim.x + threadIdx.x;
    if (i < n) out[i] = 0.0f;
}

// One direction of chamfer: for each point in X (rows), min over all points in
// Y (cols) of squared L2 distance; wave-level sum of row-mins is atomically
// accumulated (scaled) into *out. Called twice with X/Y swapped.
//
// Each wave owns one 16-row tile of X for one batch and sweeps all 16-col
// tiles of Y. Per tile: a single v_wmma_f32_16x16x4_f32 computes
//   C[r][c] = -2 * x_r . y_c + |y_c|^2        (K=4 slot carries the norms)
// then d2 = max(0, C + |x_r|^2), folded into per-lane running row-mins.
// The sweep is split into a branch-free main loop over full 16-col tiles and
// one bounds-checked tail tile, so the hot loop has no EXEC manipulation.
__global__ __launch_bounds__(256) void chamfer_dir_kernel(
    const float* __restrict__ X,   // [Bn, NX, 3]  row role
    const float* __restrict__ Y,   // [Bn, NY, 3]  col role
    float* __restrict__ out,
    int Bn, int NX, int NY, float scale)
{
    const int lane          = threadIdx.x & 31;
    const int waveInBlock   = threadIdx.x >> 5;
    const int wavesPerBlock = blockDim.x >> 5;
    const int ntilesX       = (NX + TILE - 1) / TILE;
    const int tile_id       = blockIdx.x * wavesPerBlock + waveInBlock;
    if (tile_id >= Bn * ntilesX) return;   // whole-wave exit: EXEC stays all-1s

    const int b    = tile_id / ntilesX;
    const int nt   = tile_id % ntilesX;
    const int row0 = nt * TILE;

    const int half = lane >> 4;   // 0: K=0..1 slice, 1: K=2..3 slice
    const int l16  = lane & 15;

    // ---- A operand (constant over the m-loop) ----
    // Row r = row0 + l16. A-matrix F32 16x4 layout:
    //   lanes 0-15 : VGPR0=K0, VGPR1=K1   -> (x0, x1)
    //   lanes 16-31: VGPR0=K2, VGPR1=K3   -> (x2, 1.0)
    const int r = row0 + l16;
    float x0 = 0.f, x1 = 0.f, x2c = 0.f;
    if (r < NX) {
        const float* p = X + ((size_t)b * NX + (size_t)r) * 3;
        x0 = p[0]; x1 = p[1]; x2c = p[2];
    }
    const float xsq = fmaf(x0, x0, fmaf(x1, x1, x2c * x2c));

    v2f amat;
    amat.x = half ? x2c  : x0;
    amat.y = half ? 1.0f : x1;

    // |x|^2 for the 8 C/D rows this lane will see: row = v + 8*half,
    // held by lane (v + 8*half) in lanes 0..15. Broadcast via bpermute.
    float xsqr[8];
    #pragma unroll
    for (int v = 0; v < 8; ++v)
        xsqr[v] = __shfl(xsq, v + 8 * half, 32);

    float rmin[8];
    #pragma unroll
    for (int v = 0; v < 8; ++v) rmin[v] = 3.0e38f;

    const float* Yb          = Y + (size_t)b * (size_t)NY * 3;
    const int    mtiles_full = NY / TILE;      // branch-free iterations
    const int    tail        = NY % TILE;      // leftover cols (0..15)

    // ---- Main loop: all 16 cols valid, no bounds checks, no EXEC churn ----
    for (int mt = 0; mt < mtiles_full; ++mt) {
        const int    c = mt * TILE + l16;
        const float* q = Yb + (size_t)c * 3;
        const float  y0 = q[0], y1 = q[1], y2 = q[2];
        const float  ysq = fmaf(y0, y0, fmaf(y1, y1, y2 * y2));

        // B-matrix F32 4x16 layout:
        //   lanes 0-15 : VGPR0=K0 row, VGPR1=K1 row -> (-2y0, -2y1)
        //   lanes 16-31: VGPR0=K2 row, VGPR1=K3 row -> (-2y2, |y|^2)
        v2f bmat;
        bmat.x = half ? (-2.0f * y2) : (-2.0f * y0);
        bmat.y = half ? ysq          : (-2.0f * y1);

        v8f cacc = {};
        cacc = __builtin_amdgcn_wmma_f32_16x16x4_f32(
            /*neg_a=*/false, amat, /*neg_b=*/false, bmat,
            /*c_mod=*/(short)0, cacc, /*reuse_a=*/false, /*reuse_b=*/false);

        #pragma unroll
        for (int v = 0; v < 8; ++v) {
            float d2 = fmaxf(cacc[v] + xsqr[v], 0.0f);
            rmin[v]  = fminf(rmin[v], d2);
        }
    }

    // ---- Tail tile: pad cols get |y|^2 = huge so they never win a min ----
    if (tail != 0) {
        const int c  = mtiles_full * TILE + l16;
        float y0 = 0.f, y1 = 0.f, y2 = 0.f;
        float ysq = 1.0e30f;
        if (c < NY) {
            const float* q = Yb + (size_t)c * 3;
            y0  = q[0]; y1 = q[1]; y2 = q[2];
            ysq = fmaf(y0, y0, fmaf(y1, y1, y2 * y2));
        }
        v2f bmat;
        bmat.x = half ? (-2.0f * y2) : (-2.0f * y0);
        bmat.y = half ? ysq          : (-2.0f * y1);

        v8f cacc = {};
        cacc = __builtin_amdgcn_wmma_f32_16x16x4_f32(
            false, amat, false, bmat, (short)0, cacc, false, false);

        #pragma unroll
        for (int v = 0; v < 8; ++v) {
            float d2 = fmaxf(cacc[v] + xsqr[v], 0.0f);
            rmin[v]  = fminf(rmin[v], d2);
        }
    }

    // Each lane's rmin[v] is the min over its column subset; reduce across the
    // 16 lanes of each half to get full row mins.
    #pragma unroll
    for (int v = 0; v < 8; ++v) {
        float m = rmin[v];
        m = fminf(m, __shfl_xor(m, 1, 16));
        m = fminf(m, __shfl_xor(m, 2, 16));
        m = fminf(m, __shfl_xor(m, 4, 16));
        m = fminf(m, __shfl_xor(m, 8, 16));
        rmin[v] = m;
    }

    // Sum row-mins over valid rows (rows 0..7 live in half 0, 8..15 in half 1).
    float s = 0.0f;
    #pragma unroll
    for (int v = 0; v < 8; ++v) {
        int row = row0 + v + 8 * half;
        if (row < NX) s += rmin[v];
    }
    s += __shfl_xor(s, 16, 32);   // combine the two halves

    if (lane == 0)
        __hip_atomic_fetch_add(out, s * scale,
                               __ATOMIC_RELAXED, __HIP_MEMORY_SCOPE_AGENT);
}

extern "C" void kernel_launch(void* const* d_in, const int* in_sizes, int n_in,
                              void* d_out, int out_size, void* d_ws, size_t ws_size,
                              hipStream_t stream) {
    (void)n_in; (void)d_ws; (void)ws_size; (void)in_sizes;

    const float* pred = (const float*)d_in[0];   // [4, 5000, 3]
    const float* gt   = (const float*)d_in[1];   // [4, 5000, 3]
    float* out        = (float*)d_out;

    const int B = 4, N = 5000, M = 5000;

    chamfer_zero_kernel<<<(out_size + 255) / 256, 256, 0, stream>>>(out, out_size);

    const int ntiles        = (N + TILE - 1) / TILE;          // 313
    const int total_waves   = B * ntiles;                     // 1252
    const int wavesPerBlock = 256 / 32;                       // 8
    const int grid          = (total_waves + wavesPerBlock - 1) / wavesPerBlock;

    const float scale = 1.0f / (float)(B * N);                // point+batch mean

    // cham_x: pred -> gt nearest neighbor
    chamfer_dir_kernel<<<grid, 256, 0, stream>>>(pred, gt, out, B, N, M, scale);
    // cham_y: gt -> pred nearest neighbor (same kernel, roles swapped)
    chamfer_dir_kernel<<<grid, 256, 0, stream>>>(gt, pred, out, B, M, N, scale);
}